// CrossAttention_55405078119179
// MI455X (gfx1250) — compile-verified
//
#include <hip/hip_runtime.h>
#include <hip/hip_bf16.h>

typedef __attribute__((ext_vector_type(2))) float v2f;
typedef __attribute__((ext_vector_type(8))) float v8f;

#define C_DIM 256
#define D_DIM 128
#define N_KEYS 32
#define ROWS 16384            // B*M
#define LN_EPS 1e-5f
#define DOT_SCALE 0.08838834764831843f   // 1/sqrt(128)

// workspace layout (float offsets)
#define WS_XLN 0u
#define WS_PPK (WS_XLN + 16384u*256u)          // 128*256 float2 (P packed in K-pairs)
#define WS_C0  (WS_PPK + 128u*256u*2u)         // 256
#define WS_V0  (WS_C0 + 256u)                  // 256
#define WS_S0  (WS_V0 + 256u)                  // 1 (+pad)
#define WS_G   (WS_S0 + 256u)                  // 16384*256
#define WS_T   (WS_G + 16384u*256u)            // 16384
#define WS_GS  (WS_T + 16384u)                 // 16384
#define WS_PG  (WS_GS + 16384u)                // 16384*16 per-tile rowsum(g) partials
#define WS_PBW (WS_PG + 16384u*16u)            // 16384*16 per-tile rowsum(beta*w)+bkq partials

// ---------------------------------------------------------------------------
// Kernel 0: P = Wq @ Wk^T packed as K-pairs; c0 = Wk@bq; v0 = Wq@bk; s0 = bk.bq
// grid 128, block 256.  Ppk[p*256 + c'] = {P[2p][c'], P[2p+1][c']}
// ---------------------------------------------------------------------------
__global__ void __launch_bounds__(256)
prep_kernel(const float* __restrict__ Wq, const float* __restrict__ Wk,
            const float* __restrict__ bq, const float* __restrict__ bk,
            float* __restrict__ ws) {
  const int cp = threadIdx.x;   // column c' of P
  const int p  = blockIdx.x;    // K-pair index
  const float* wq0 = Wq + (2 * p) * D_DIM;
  const float* wq1 = wq0 + D_DIM;
  const float* wkr = Wk + cp * D_DIM;
  float a0 = 0.f, a1 = 0.f;
  #pragma unroll 4
  for (int d = 0; d < D_DIM; ++d) {
    const float kv = wkr[d];
    a0 += wq0[d] * kv;
    a1 += wq1[d] * kv;
  }
  float2 o; o.x = a0; o.y = a1;
  ((float2*)(ws + WS_PPK))[p * C_DIM + cp] = o;

  if (blockIdx.x == 0) {
    float c0 = 0.f, v0 = 0.f;
    for (int d = 0; d < D_DIM; ++d) {
      c0 += wkr[d] * bq[d];
      v0 += Wq[cp * D_DIM + d] * bk[d];
    }
    ws[WS_C0 + cp] = c0;
    ws[WS_V0 + cp] = v0;
    if (cp == 0) {
      float s = 0.f;
      for (int d = 0; d < D_DIM; ++d) s += bk[d] * bq[d];
      ws[WS_S0] = s;
    }
  }
}

// ---------------------------------------------------------------------------
// Kernel 1: xln = LayerNorm(x) * q_gamma + q_beta.  grid ROWS, block 256.
// ---------------------------------------------------------------------------
__global__ void __launch_bounds__(256)
lnx_kernel(const float* __restrict__ x, const float* __restrict__ qg,
           const float* __restrict__ qb, float* __restrict__ ws) {
  const int row = blockIdx.x;
  const int tid = threadIdx.x;
  const float v = x[(size_t)row * C_DIM + tid];
  float s = v, s2 = v * v;
  #pragma unroll
  for (int off = 16; off > 0; off >>= 1) {
    s  += __shfl_xor(s,  off);
    s2 += __shfl_xor(s2, off);
  }
  __shared__ float ls[8], ls2[8];
  const int w = tid >> 5;
  if ((tid & 31) == 0) { ls[w] = s; ls2[w] = s2; }
  __syncthreads();
  float S = 0.f, S2 = 0.f;
  #pragma unroll
  for (int i = 0; i < 8; ++i) { S += ls[i]; S2 += ls2[i]; }
  const float mu  = S * (1.0f / C_DIM);
  const float var = S2 * (1.0f / C_DIM) - mu * mu;
  const float inv = rsqrtf(var + LN_EPS);
  ws[WS_XLN + (size_t)row * C_DIM + tid] = (v - mu) * inv * qg[tid] + qb[tid];
}

// ---------------------------------------------------------------------------
// Kernel 2: WMMA f32 GEMM, one wave per 16x16 tile (16384 waves -> full chip).
// K loop in 8 chunks of 8: clause of 16 loads, then 8 back-to-back WMMAs on
// the accumulator chain. Per-tile row-sum partials go to WS_PG/WS_PBW
// (deterministic; no float atomics). bkq handled only by the nt==0 wave in a
// separate uniform loop (no branch in the WMMA loop).
// grid 4096, block 128 (4 waves).
// ---------------------------------------------------------------------------
__global__ void __launch_bounds__(128)
gemm_kernel(const float* __restrict__ kgm, const float* __restrict__ kbt,
            float* __restrict__ ws) {
  const float*  xln = ws + WS_XLN;
  const float2* ppk = (const float2*)(ws + WS_PPK);
  const float*  c0  = ws + WS_C0;
  const float*  v0  = ws + WS_V0;
  const float   s0v = ws[WS_S0];
  float* g   = ws + WS_G;
  float* pg  = ws + WS_PG;
  float* pbw = ws + WS_PBW;

  const int lane   = threadIdx.x & 31;
  const int wvid   = blockIdx.x * 4 + (threadIdx.x >> 5);   // 0..16383
  const int tile_m = (wvid >> 4) * 16;
  const int nt     = wvid & 15;                             // column tile
  const int hi     = lane >> 4;      // 0: K%4 in {0,1}, 1: K%4 in {2,3}
  const int lo     = lane & 15;
  const int arow   = tile_m + lo;    // A-fragment row for this lane
  const int col    = nt * 16 + lo;   // B/D column for this lane

  const float*  ap = xln + (size_t)arow * C_DIM + hi * 2;
  const float2* bp = ppk + (size_t)hi * C_DIM + col;

  v8f acc = {0.f, 0.f, 0.f, 0.f, 0.f, 0.f, 0.f, 0.f};
  #pragma unroll 1
  for (int kc = 0; kc < 8; ++kc) {          // 8 chunks x 8 K-steps of 4
    v2f    af[8];
    float2 bf[8];
    #pragma unroll
    for (int i = 0; i < 8; ++i) {           // one clause of 16 b64 loads
      const int ks = kc * 8 + i;
      af[i] = *(const v2f*)(ap + ks * 4);
      bf[i] = bp[(size_t)ks * 2 * C_DIM];
    }
    #pragma unroll
    for (int i = 0; i < 8; ++i) {           // 8 chained WMMAs
      v2f bv; bv[0] = bf[i].x; bv[1] = bf[i].y;
      acc = __builtin_amdgcn_wmma_f32_16x16x4_f32(
          /*neg_a=*/false, af[i], /*neg_b=*/false, bv,
          /*c_mod=*/(short)0, acc, /*reuse_a=*/false, /*reuse_b=*/false);
    }
  }

  const float c0v = c0[col];
  const float kgv = kgm[col];
  const float kbv = kbt[col];
  float rs_g[8], rs_bw[8];
  // C/D layout: VGPR j -> row tile_m + j (lanes 0-15) / tile_m + j + 8 (16-31)
  #pragma unroll
  for (int j = 0; j < 8; ++j) {
    const int row = tile_m + j + hi * 8;
    const float wvv = acc[j] + c0v;
    const float gvv = kgv * wvv;
    g[(size_t)row * C_DIM + col] = gvv;
    rs_g[j]  = gvv;
    rs_bw[j] = kbv * wvv;
  }
  #pragma unroll
  for (int j = 0; j < 8; ++j) {
    #pragma unroll
    for (int off = 1; off < 16; off <<= 1) {     // stays within 16-lane halves
      rs_g[j]  += __shfl_xor(rs_g[j],  off);
      rs_bw[j] += __shfl_xor(rs_bw[j], off);
    }
  }

  // bkq = xln[row].v0 + s0, computed once per row by the nt==0 wave only
  float bkq = 0.f;
  if (nt == 0) {
    #pragma unroll 4
    for (int ks = 0; ks < 64; ++ks) {
      const int kb = ks * 4 + hi * 2;
      const v2f a  = *(const v2f*)(xln + (size_t)arow * C_DIM + kb);
      const v2f vv = *(const v2f*)(v0 + kb);
      bkq += a[0] * vv[0] + a[1] * vv[1];
    }
    bkq += __shfl_xor(bkq, 16);   // lane l: bkq for row tile_m+(l&15)
    bkq += s0v;
  }

  #pragma unroll
  for (int j = 0; j < 8; ++j) {
    const float b0 = __shfl(bkq, j);        // bkq for row tile_m+j   (0 if nt!=0)
    const float b1 = __shfl(bkq, j + 8);    // bkq for row tile_m+j+8 (0 if nt!=0)
    if (lane == 0) {
      pg [(size_t)(tile_m + j) * 16 + nt] = rs_g[j];
      pbw[(size_t)(tile_m + j) * 16 + nt] = rs_bw[j] + b0;
    }
    if (lane == 16) {
      pg [(size_t)(tile_m + j + 8) * 16 + nt] = rs_g[j];
      pbw[(size_t)(tile_m + j + 8) * 16 + nt] = rs_bw[j] + b1;
    }
  }
}

// ---------------------------------------------------------------------------
// Kernel 3: deterministic reduction of the 16 per-tile partials per row.
// grid 64, block 256.
// ---------------------------------------------------------------------------
__global__ void __launch_bounds__(256)
reduce_kernel(float* __restrict__ ws) {
  const int r = blockIdx.x * 256 + threadIdx.x;   // 0..16383
  const float* pg  = ws + WS_PG  + (size_t)r * 16;
  const float* pbw = ws + WS_PBW + (size_t)r * 16;
  float a = 0.f, b = 0.f;
  #pragma unroll
  for (int i = 0; i < 16; ++i) { a += pg[i]; b += pbw[i]; }
  ws[WS_GS + r] = a;
  ws[WS_T  + r] = b;
}

// ---------------------------------------------------------------------------
// Kernel 4: the 512-MiB bandwidth kernel. One wave per (b,m).
// 4 y-rows per iteration via 8-lane subgroups: coalesced b128 loads,
// only 10 cross-lane ops per 4 rows. Softmax over 32 lanes fused with z.
// grid 2048, block 256 (8 waves).
// ---------------------------------------------------------------------------
__global__ void __launch_bounds__(256)
attn_kernel(const float* __restrict__ y, const float* __restrict__ z,
            const float* __restrict__ ws, float* __restrict__ out) {
  const float* g  = ws + WS_G;
  const float* t  = ws + WS_T;
  const float* gs = ws + WS_GS;

  const int lane = threadIdx.x & 31;
  const int r    = blockIdx.x * 8 + (threadIdx.x >> 5);   // b*M + m
  const int gr   = lane >> 3;    // row subgroup 0..3
  const int sl   = lane & 7;     // sublane within group
  const float* yrow = y + (size_t)r * (N_KEYS * C_DIM);
  const float* grow = g + (size_t)r * C_DIM;

  // g fragment: 32 columns per lane (cols sl*4 + c*32 .. +3); shared across rows
  float4 gf[8];
  #pragma unroll
  for (int c = 0; c < 8; ++c)
    gf[c] = *(const float4*)(grow + c * 32 + sl * 4);
  const float tv  = t[r];
  const float gsv = gs[r];

  float myd = 0.f;
  #pragma unroll 2
  for (int it = 0; it < 8; ++it) {
    const float* yp = yrow + (size_t)(4 * it + gr) * C_DIM + sl * 4;
    __builtin_prefetch(yp + 4 * C_DIM, 0, 1);   // speculative: safe past end
    float s = 0.f, s2 = 0.f, d = 0.f;
    #pragma unroll
    for (int c = 0; c < 8; ++c) {
      const float4 yv = *(const float4*)(yp + c * 32);
      s  += yv.x + yv.y + yv.z + yv.w;
      s2 += yv.x*yv.x + yv.y*yv.y + yv.z*yv.z + yv.w*yv.w;
      d  += yv.x*gf[c].x + yv.y*gf[c].y + yv.z*gf[c].z + yv.w*gf[c].w;
    }
    #pragma unroll
    for (int off = 1; off < 8; off <<= 1) {     // reduce within 8-lane group
      s  += __shfl_xor(s,  off);
      s2 += __shfl_xor(s2, off);
      d  += __shfl_xor(d,  off);
    }
    const float mu  = s * (1.0f / C_DIM);
    const float var = s2 * (1.0f / C_DIM) - mu * mu;
    const float inv = rsqrtf(var + LN_EPS);
    const float dn  = DOT_SCALE * (inv * (d - mu * gsv) + tv);  // row 4*it+gr
    // scatter: lane L wants row 4*it+(L&3), held by group (L&3) (lane (L&3)*8)
    const float got = __shfl(dn, (lane & 3) << 3);
    if ((lane >> 2) == it) myd = got;
  }

  // softmax over 32 lanes, fused with the z-weighted average
  float mx = myd;
  #pragma unroll
  for (int off = 16; off > 0; off >>= 1) mx = fmaxf(mx, __shfl_xor(mx, off));
  const float e  = __expf(myd - mx);
  const float zv = z[(size_t)r * N_KEYS + lane];
  float num = e * zv, den = e;
  #pragma unroll
  for (int off = 16; off > 0; off >>= 1) {
    num += __shfl_xor(num, off);
    den += __shfl_xor(den, off);
  }
  if (lane == 0) out[r] = num / den;
}

// ---------------------------------------------------------------------------
extern "C" void kernel_launch(void* const* d_in, const int* in_sizes, int n_in,
                              void* d_out, int out_size, void* d_ws, size_t ws_size,
                              hipStream_t stream) {
  const float* x       = (const float*)d_in[0];
  const float* y       = (const float*)d_in[1];
  const float* z       = (const float*)d_in[2];
  const float* q_gamma = (const float*)d_in[3];
  const float* q_beta  = (const float*)d_in[4];
  const float* Wq      = (const float*)d_in[5];
  const float* bq      = (const float*)d_in[6];
  const float* k_gamma = (const float*)d_in[7];
  const float* k_beta  = (const float*)d_in[8];
  const float* Wk      = (const float*)d_in[9];
  const float* bk      = (const float*)d_in[10];
  float* ws  = (float*)d_ws;
  float* out = (float*)d_out;

  prep_kernel  <<<128,  256, 0, stream>>>(Wq, Wk, bq, bk, ws);
  lnx_kernel   <<<ROWS, 256, 0, stream>>>(x, q_gamma, q_beta, ws);
  gemm_kernel  <<<4096, 128, 0, stream>>>(k_gamma, k_beta, ws);
  reduce_kernel<<<64,   256, 0, stream>>>(ws);
  attn_kernel  <<<2048, 256, 0, stream>>>(y, z, ws, out);
}